// GraphVAE_37005438222392
// MI455X (gfx1250) — compile-verified
//
#include <hip/hip_runtime.h>
#include <hip/hip_bf16.h>

typedef float v2f __attribute__((ext_vector_type(2)));
typedef float v8f __attribute__((ext_vector_type(8)));

static constexpr int Nn = 1024, Ff = 1000, Hh = 256, LSc = 64, LYc = 64, NLc = 8;
static constexpr int NEc = Nn * (Nn + 1) / 2;   // 524800

#define WMMA_F32(a, b, c) \
    __builtin_amdgcn_wmma_f32_16x16x4_f32(false, (a), false, (b), (short)0, (c), false, false)

// ---------------- WMMA fp32 GEMM: C[M,N] = A[M,K](lda) @ B[K,N] ----------------
// One wave -> 16x64 output tile (4 accumulators sharing one A fragment).
// All loads are branchless (index clamp); K remainder handled by a uniform
// scalar branch that zero-masks the A fragment. EXEC stays all-ones throughout.
__global__ void k_gemm_wmma(const float* __restrict__ A, int lda,
                            const float* __restrict__ B,
                            float* __restrict__ C,
                            int M, int N, int K) {
    const int lane  = threadIdx.x & 31;
    const int wave  = threadIdx.x >> 5;
    const int tileM = blockIdx.y * 64 + wave * 16;
    const int tileN = blockIdx.x * 64;
    const int half  = lane >> 4;          // 0 or 1
    const int l15   = lane & 15;
    const int khalf = half << 1;          // 0 or 2

    int mr = tileM + l15;  mr = (mr < M) ? mr : (M - 1);     // clamp (rows >=M never stored)
    const int nc0 = min(tileN + l15,      N - 1);
    const int nc1 = min(tileN + 16 + l15, N - 1);
    const int nc2 = min(tileN + 32 + l15, N - 1);
    const int nc3 = min(tileN + 48 + l15, N - 1);

    const float* __restrict__ Arow = A + (size_t)mr * lda;
    const float* __restrict__ B0   = B + (size_t)khalf * N;
    const float* __restrict__ B1   = B0 + N;
    const int K4 = K & ~3;

    v8f acc0 = {}, acc1 = {}, acc2 = {}, acc3 = {};
    #pragma unroll 2
    for (int k = 0; k < K4; k += 4) {
        v2f a;
        a.x = Arow[k + khalf];
        a.y = Arow[k + khalf + 1];
        v2f b;
        b.x = B0[nc0]; b.y = B1[nc0];  acc0 = WMMA_F32(a, b, acc0);
        b.x = B0[nc1]; b.y = B1[nc1];  acc1 = WMMA_F32(a, b, acc1);
        b.x = B0[nc2]; b.y = B1[nc2];  acc2 = WMMA_F32(a, b, acc2);
        b.x = B0[nc3]; b.y = B1[nc3];  acc3 = WMMA_F32(a, b, acc3);
        B0 += (size_t)4 * N;
        B1 += (size_t)4 * N;
    }
    if (K4 < K) {                         // uniform branch: only K=1001 takes it
        int ka = K4 + khalf;
        int k0 = (ka     < K) ? ka     : K - 1;
        int k1 = (ka + 1 < K) ? ka + 1 : K - 1;
        float m0 = (ka     < K) ? 1.0f : 0.0f;
        float m1 = (ka + 1 < K) ? 1.0f : 0.0f;
        v2f a;
        a.x = Arow[k0] * m0;
        a.y = Arow[k1] * m1;
        const float* Bt0 = B + (size_t)k0 * N;
        const float* Bt1 = B + (size_t)k1 * N;
        v2f b;
        b.x = Bt0[nc0]; b.y = Bt1[nc0];  acc0 = WMMA_F32(a, b, acc0);
        b.x = Bt0[nc1]; b.y = Bt1[nc1];  acc1 = WMMA_F32(a, b, acc1);
        b.x = Bt0[nc2]; b.y = Bt1[nc2];  acc2 = WMMA_F32(a, b, acc2);
        b.x = Bt0[nc3]; b.y = Bt1[nc3];  acc3 = WMMA_F32(a, b, acc3);
    }

    const int mbase = tileM + (half << 3);
    v8f accs[4] = {acc0, acc1, acc2, acc3};
    #pragma unroll
    for (int t = 0; t < 4; ++t) {
        int c = tileN + 16 * t + l15;
        if (c < N) {
            #pragma unroll
            for (int j = 0; j < 8; ++j) {
                int r = mbase + j;
                if (r < M) C[(size_t)r * N + c] = accs[t][j];
            }
        }
    }
}

// ---------------- GCN normalization ----------------
__global__ void k_fill(float* p, float v, int n) {
    int t = blockIdx.x * blockDim.x + threadIdx.x;
    if (t < n) p[t] = v;
}
__global__ void k_deg(const int* __restrict__ dst, float* deg, int E) {
    int t = blockIdx.x * blockDim.x + threadIdx.x;
    if (t < E) unsafeAtomicAdd(&deg[dst[t]], 1.0f);
}
__global__ void k_dis(float* d, int n) {   // in-place: rsqrt(max(deg,1))
    int t = blockIdx.x * blockDim.x + threadIdx.x;
    if (t < n) d[t] = rsqrtf(fmaxf(d[t], 1.0f));
}
__global__ void k_edge_norm(const int* __restrict__ src, const int* __restrict__ dst,
                            const float* __restrict__ dis, float* nrm, int E) {
    int t = blockIdx.x * blockDim.x + threadIdx.x;
    if (t < E) nrm[t] = dis[src[t]] * dis[dst[t]];
}

// out[i,:] = bias + h[i,:]*dis[i]^2   (self loop contribution)
__global__ void k_gcn_init(const float* __restrict__ h, const float* __restrict__ bias,
                           const float* __restrict__ dis, float* out, int n, int W) {
    int t = blockIdx.x * blockDim.x + threadIdx.x;
    if (t >= n * W) return;
    int r = t / W;
    float d = dis[r];
    out[t] = bias[t % W] + h[t] * d * d;
}
// out[dst,:] += h[src,:]*norm[e]   (hardware fp32 atomics, coalesced per edge row)
__global__ void k_gcn_scatter(const float* __restrict__ h, const int* __restrict__ src,
                              const int* __restrict__ dst, const float* __restrict__ nrm,
                              float* out, int E, int W) {
    long long t = (long long)blockIdx.x * blockDim.x + threadIdx.x;
    if (t >= (long long)E * W) return;
    int e = (int)(t / W), j = (int)(t % W);
    unsafeAtomicAdd(&out[(size_t)dst[e] * W + j], h[(size_t)src[e] * W + j] * nrm[e]);
}

// ---------------- elementwise ----------------
__global__ void k_relu(float* p, int n) {
    int t = blockIdx.x * blockDim.x + threadIdx.x;
    if (t < n) p[t] = fmaxf(p[t], 0.0f);
}
__global__ void k_sig_relu(const float* in, float* out, int n) {  // sigmoid(relu(x))
    int t = blockIdx.x * blockDim.x + threadIdx.x;
    if (t < n) { float x = fmaxf(in[t], 0.0f); out[t] = 1.0f / (1.0f + __expf(-x)); }
}
__global__ void k_copy(const float* __restrict__ src, int sstride, int soff,
                       float* __restrict__ dst, int dstride, int doff,
                       int rows, int width) {
    int t = blockIdx.x * blockDim.x + threadIdx.x;
    if (t >= rows * width) return;
    int r = t / width, c = t % width;
    dst[(size_t)r * dstride + doff + c] = src[(size_t)r * sstride + soff + c];
}
__global__ void k_reparam(const float* __restrict__ eps, const float* __restrict__ mu,
                          const float* __restrict__ lv, float* latent,
                          int n, int L, int stride, int off) {
    int t = blockIdx.x * blockDim.x + threadIdx.x;
    if (t >= n * L) return;
    int r = t / L, c = t % L;
    latent[(size_t)r * stride + off + c] = eps[t] * __expf(0.5f * lv[t]) + mu[t];
}
__global__ void k_softmax8(const float* __restrict__ in, float* out, int rows) {
    int r = blockIdx.x * blockDim.x + threadIdx.x;
    if (r >= rows) return;
    float v[8], m = -1e30f;
    #pragma unroll
    for (int j = 0; j < 8; ++j) { v[j] = in[r * 8 + j]; m = fmaxf(m, v[j]); }
    float s = 0.0f;
    #pragma unroll
    for (int j = 0; j < 8; ++j) { v[j] = __expf(v[j] - m); s += v[j]; }
    float inv = 1.0f / s;
    #pragma unroll
    for (int j = 0; j < 8; ++j) out[r * 8 + j] = v[j] * inv;
}

// ---------------- A decoder matvecs (bandwidth-bound streaming) ----------------
__global__ void k_matvec1(const float* __restrict__ feat, const float* __restrict__ W,
                          float* hidden) {
    int t = threadIdx.x;                 // 256 threads, 2 outputs each (O=512)
    int k0 = blockIdx.x * 512;
    float a0 = 0.0f, a1 = 0.0f;
    for (int k = 0; k < 512; ++k) {
        float f = feat[k0 + k];
        const float* row = W + (size_t)(k0 + k) * 512;
        a0 += f * row[t];
        a1 += f * row[t + 256];
    }
    unsafeAtomicAdd(&hidden[t], a0);
    unsafeAtomicAdd(&hidden[t + 256], a1);
}
__global__ void k_matvec2(const float* __restrict__ hidden, const float* __restrict__ W,
                          const float* __restrict__ b, float* l, int ne) {
    __shared__ float hs[512];
    for (int i = threadIdx.x; i < 512; i += blockDim.x) hs[i] = hidden[i];
    __syncthreads();
    int e = blockIdx.x * blockDim.x + threadIdx.x;
    if (e >= ne) return;
    float acc = b[e];
    for (int k = 0; k < 512; ++k) acc += hs[k] * W[(size_t)k * ne + e];
    l[e] = acc;
}
// A[i,j]=A[j,i]=sigmoid(l[triu_idx(i,j)]); covers full matrix incl. diagonal.
__global__ void k_assembleA(const float* __restrict__ l, float* A, int n) {
    int i = blockIdx.x;
    int base = i * n - (i * (i - 1)) / 2 - i;   // idx(i,j) = base + j
    for (int j = i + threadIdx.x; j < n; j += blockDim.x) {
        float s = 1.0f / (1.0f + __expf(-l[base + j]));
        A[(size_t)i * n + j] = s;
        A[(size_t)j * n + i] = s;
    }
}

extern "C" void kernel_launch(void* const* d_in, const int* in_sizes, int n_in,
                              void* d_out, int out_size, void* d_ws, size_t ws_size,
                              hipStream_t stream) {
    const float* X     = (const float*)d_in[0];
    const float* Y     = (const float*)d_in[1];
    const float* epsS  = (const float*)d_in[2];
    const float* epsY  = (const float*)d_in[3];
    const float* Wse1  = (const float*)d_in[4];  const float* bse1  = (const float*)d_in[5];
    const float* WseMu = (const float*)d_in[6];  const float* bseMu = (const float*)d_in[7];
    const float* WseLs = (const float*)d_in[8];  const float* bseLs = (const float*)d_in[9];
    const float* Wye1  = (const float*)d_in[10]; const float* bye1  = (const float*)d_in[11];
    const float* WyeMu = (const float*)d_in[12]; const float* byeMu = (const float*)d_in[13];
    const float* WyeLs = (const float*)d_in[14]; const float* byeLs = (const float*)d_in[15];
    const float* Wsd1  = (const float*)d_in[16]; const float* bsd1  = (const float*)d_in[17];
    const float* Wsd2  = (const float*)d_in[18]; const float* bsd2  = (const float*)d_in[19];
    const float* Wyp1  = (const float*)d_in[20]; const float* byp1  = (const float*)d_in[21];
    const float* Wyp2  = (const float*)d_in[22]; const float* byp2  = (const float*)d_in[23];
    const float* Wa1   = (const float*)d_in[24]; const float* ba1   = (const float*)d_in[25];
    const float* Wa2   = (const float*)d_in[26]; const float* ba2   = (const float*)d_in[27];
    const float* Wxd1  = (const float*)d_in[28]; const float* bxd1  = (const float*)d_in[29];
    const float* Wxd2  = (const float*)d_in[30]; const float* bxd2  = (const float*)d_in[31];
    const float* Wyd1  = (const float*)d_in[32]; const float* byd1  = (const float*)d_in[33];
    const float* Wyd2  = (const float*)d_in[34]; const float* byd2  = (const float*)d_in[35];
    const int*   eidx  = (const int*)d_in[36];
    const int E = in_sizes[36] / 2;
    const int* srcI = eidx;
    const int* dstI = eidx + E;

    float* out = (float*)d_out;
    const size_t o_Xn = 0, o_l = 1024000, o_A = 1548800, o_Yp = 2597376,
                 o_Yn = 2605568, o_Sh = 2613760, o_muS = 2614784,
                 o_lvS = 2680320, o_muY = 2745856, o_lvY = 2811392;

    float* ws     = (float*)d_ws;
    float* dis    = ws;                        // 1024
    float* enorm  = ws + 1024;                 // 32768
    float* Xcat   = ws + 33792;                // 1024*1001
    float* catYX  = ws + 1058816;              // 1024*1064
    float* latent = ws + 2148352;              // 1024*128
    float* bufH   = ws + 2279424;              // 1024*256
    float* bufHY  = ws + 2541568;              // 1024*256
    float* tmpA   = ws + 2803712;              // 1024*1000 (gemm output)
    float* tmpB   = ws + 3827712;              // 1024*512
    float* sbuf1  = ws + 4352000;              // 1024
    float* sbuf8  = ws + 4353024;              // 1024*8
    float* ahid   = ws + 4361216;              // 512

    auto gemm = [&](const float* A, int lda, const float* B, float* C, int M, int Ncols, int K) {
        dim3 g((Ncols + 63) / 64, (M + 63) / 64);
        k_gemm_wmma<<<g, 128, 0, stream>>>(A, lda, B, C, M, Ncols, K);
    };
    auto gcn = [&](const float* x, int lda, int K, const float* W, const float* b,
                   int Wd, float* o) {
        gemm(x, lda, W, tmpA, Nn, Wd, K);
        int tot = Nn * Wd;
        k_gcn_init<<<(tot + 255) / 256, 256, 0, stream>>>(tmpA, b, dis, o, Nn, Wd);
        long long et = (long long)E * Wd;
        k_gcn_scatter<<<(unsigned)((et + 255) / 256), 256, 0, stream>>>(tmpA, srcI, dstI, enorm, o, E, Wd);
    };
    auto relu = [&](float* p, int n) { k_relu<<<(n + 255) / 256, 256, 0, stream>>>(p, n); };

    // ---- GCN normalization coefficients ----
    k_fill<<<(Nn + 255) / 256, 256, 0, stream>>>(dis, 1.0f, Nn);     // self loops
    k_deg<<<(E + 255) / 256, 256, 0, stream>>>(dstI, dis, E);
    k_dis<<<(Nn + 255) / 256, 256, 0, stream>>>(dis, Nn);
    k_edge_norm<<<(E + 255) / 256, 256, 0, stream>>>(srcI, dstI, dis, enorm, E);

    // ---- Xcat = [X | Y] ----
    k_copy<<<(Nn * Ff + 255) / 256, 256, 0, stream>>>(X, Ff, 0, Xcat, Ff + 1, 0, Nn, Ff);
    k_copy<<<(Nn + 255) / 256, 256, 0, stream>>>(Y, 1, 0, Xcat, Ff + 1, Ff, Nn, 1);

    // ---- U_S encoder ----
    gcn(X, Ff, Ff, Wse1, bse1, Hh, bufH);          relu(bufH, Nn * Hh);
    gcn(bufH, Hh, Hh, WseMu, bseMu, LSc, out + o_muS);
    gcn(bufH, Hh, Hh, WseLs, bseLs, LSc, out + o_lvS);

    // ---- U_Y encoder ----
    gcn(Xcat, Ff + 1, Ff + 1, Wye1, bye1, Hh, bufHY);  relu(bufHY, Nn * Hh);
    gcn(bufHY, Hh, Hh, WyeMu, byeMu, LYc, out + o_muY);
    gcn(bufHY, Hh, Hh, WyeLs, byeLs, LYc, out + o_lvY);

    // ---- reparameterize into latent = [u_S | u_Y] (stride 128) ----
    k_reparam<<<(Nn * LSc + 255) / 256, 256, 0, stream>>>(epsS, out + o_muS, out + o_lvS, latent, Nn, LSc, 128, 0);
    k_reparam<<<(Nn * LYc + 255) / 256, 256, 0, stream>>>(epsY, out + o_muY, out + o_lvY, latent, Nn, LYc, 128, 64);

    // ---- S decoder: sigmoid(relu(gcn(relu(gcn(u_S))))) ----
    gcn(latent, 128, LSc, Wsd1, bsd1, Hh, tmpB);   relu(tmpB, Nn * Hh);
    gcn(tmpB, Hh, Hh, Wsd2, bsd2, 1, sbuf1);
    k_sig_relu<<<(Nn + 255) / 256, 256, 0, stream>>>(sbuf1, out + o_Sh, Nn);

    // ---- Y' decoder: softmax(gcn(gcn(X))) ----
    gcn(X, Ff, Ff, Wyp1, byp1, 512, tmpB);
    gcn(tmpB, 512, 512, Wyp2, byp2, NLc, sbuf8);
    k_softmax8<<<(Nn + 255) / 256, 256, 0, stream>>>(sbuf8, out + o_Yp, Nn);

    // ---- A decoder: two streaming matvecs (1.34 GB total, HBM-roofline bound) ----
    hipMemcpyAsync(ahid, ba1, 512 * sizeof(float), hipMemcpyDeviceToDevice, stream);
    k_matvec1<<<256, 256, 0, stream>>>(latent, Wa1, ahid);   // 131072 = 256*512 K-chunks
    k_matvec2<<<(NEc + 255) / 256, 256, 0, stream>>>(ahid, Wa2, ba2, out + o_l, NEc);
    k_assembleA<<<Nn, 256, 0, stream>>>(out + o_l, out + o_A, Nn);

    // ---- X decoder ----
    gcn(latent, 128, 128, Wxd1, bxd1, Hh, tmpB);
    gcn(tmpB, Hh, Hh, Wxd2, bxd2, Ff, out + o_Xn);

    // ---- Y decoder: softmax(gcn(gcn([u_Y | X]))) ----
    k_copy<<<(Nn * LYc + 255) / 256, 256, 0, stream>>>(latent, 128, 64, catYX, LYc + Ff, 0, Nn, LYc);
    k_copy<<<(Nn * Ff + 255) / 256, 256, 0, stream>>>(X, Ff, 0, catYX, LYc + Ff, LYc, Nn, Ff);
    gcn(catYX, LYc + Ff, LYc + Ff, Wyd1, byd1, 512, tmpB);
    gcn(tmpB, 512, 512, Wyd2, byd2, NLc, sbuf8);
    k_softmax8<<<(Nn + 255) / 256, 256, 0, stream>>>(sbuf8, out + o_Yn, Nn);
}